// InfoNCEEuclidean_73667279061520
// MI455X (gfx1250) — compile-verified
//
#include <hip/hip_runtime.h>

// ---------------------------------------------------------------------------
// InfoNCE-Euclidean loss, fused. features: 8192 x 128 f32.
// a = rows [0,4096), b = rows [4096,8192).
// Need: rowsum(sim_aa), rowsum(sim_bb), rowsum(sim_ab), colsum(sim_ab),
//       trace(sim_ab);  sim = 1/(1+sqrt(max(|x|^2+|y|^2-2 x.y, 0))).
// GEMM core: V_WMMA_F32_16X16X4_F32 (exact f32, matches reference).
// Tile staging: TDM tensor_load_to_lds with LDS row padding (bank-conflict-free).
// ---------------------------------------------------------------------------

#define BATCH 4096
#define DIMK  128
#define TILE  128          // block tile (rows of X and of Y)
#define NT    (BATCH/TILE) // 32 tiles per dimension
#define LROW  132          // padded LDS row stride (floats) = 128 + 4 (TDM pad)
#define NTRI  (NT*(NT+1)/2)   // 528 upper-tri tile pairs

typedef __attribute__((ext_vector_type(2))) float v2f;
typedef __attribute__((ext_vector_type(8))) float v8f;
typedef unsigned int v4u __attribute__((ext_vector_type(4)));
typedef int          v8i __attribute__((ext_vector_type(8)));
typedef int          v4i __attribute__((ext_vector_type(4)));

// ---------------- zero accumulators ----------------
__global__ void zero_ws_kernel(float* __restrict__ p, int n) {
    for (int i = blockIdx.x * blockDim.x + threadIdx.x; i < n;
         i += gridDim.x * blockDim.x)
        p[i] = 0.0f;
}

// ---------------- per-row squared norms ----------------
// one wave per row; 32 lanes x float4 = 128 floats
__global__ void sqnorm_kernel(const float* __restrict__ feats,
                              float* __restrict__ sq) {
    const int wave = threadIdx.x >> 5;
    const int lane = threadIdx.x & 31;
    const int row  = blockIdx.x * 8 + wave;          // grid = 1024 -> 8192 rows
    const float4 v = ((const float4*)(feats + (size_t)row * DIMK))[lane];
    float s = v.x * v.x + v.y * v.y + v.z * v.z + v.w * v.w;
    s += __shfl_xor(s, 1);  s += __shfl_xor(s, 2);
    s += __shfl_xor(s, 4);  s += __shfl_xor(s, 8);
    s += __shfl_xor(s, 16);
    if (lane == 0) sq[row] = s;
}

// ---------------- TDM: DMA one 128x128 f32 tile global -> LDS ----------------
// Pads LDS destination by 4 DWORDs after every 128 DWORDs (row stride = 132).
__device__ __forceinline__ void tdm_load_tile(const float* gsrc, unsigned lds_off) {
    const unsigned long long ga = (unsigned long long)(uintptr_t)gsrc;
    v4u g0;
    g0[0] = 1u;                                   // count=1, no gather
    g0[1] = lds_off;                              // LDS byte address
    g0[2] = (unsigned)ga;                         // global_addr[31:0]
    g0[3] = (unsigned)((ga >> 32) & 0x01FFFFFFu)  // global_addr[56:32]
          | (2u << 30);                           // type=2 ("image")
    v8i g1;
    g1[0] = (int)((2u << 16)        // data_size = 4 bytes
                | (1u << 20)        // pad_enable
                | (6u << 22)        // pad_interval: 2^(6+1)=128 DWORDs
                | (3u << 25));      // pad_amount: 3+1 = 4 DWORDs
    g1[1] = (int)(128u << 16);      // [15:0] barrier addr=0; [31:16] tensor_dim0 lo = 128
    g1[2] = (int)(128u << 16);      // [15:0] tensor_dim0 hi=0; [31:16] tensor_dim1 lo = 128
    g1[3] = (int)(128u << 16);      // [15:0] tensor_dim1 hi=0; [31:16] tile_dim0 = 128
    g1[4] = (int)128u;              // [15:0] tile_dim1 = 128; [31:16] tile_dim2 = 0
    g1[5] = (int)128u;              // tensor_dim0_stride[31:0] = 128 elements
    g1[6] = 0;                      // stride hi, tensor_dim1_stride lo
    g1[7] = 0;
    const v4i z4 = {0, 0, 0, 0};
#if __clang_major__ >= 23
    const v8i z8 = {0, 0, 0, 0, 0, 0, 0, 0};
    __builtin_amdgcn_tensor_load_to_lds(g0, g1, z4, z4, z8, 0);
#else
    __builtin_amdgcn_tensor_load_to_lds(g0, g1, z4, z4, 0);
#endif
}

// ---------------- main fused distance-GEMM tile kernel ----------------
__global__ __launch_bounds__(256)
void tile_kernel(const float* __restrict__ feats,
                 const float* __restrict__ sq,
                 float* __restrict__ saa,    // rowsum sim_aa   [4096]
                 float* __restrict__ sbb,    // rowsum sim_bb   [4096]
                 float* __restrict__ sabr,   // rowsum sim_ab   [4096]
                 float* __restrict__ sabc,   // colsum sim_ab   [4096]
                 float* __restrict__ trace)  // trace sim_ab    [1]
{
    const float* __restrict__ a = feats;
    const float* __restrict__ b = feats + (size_t)BATCH * DIMK;

    // ---- decode block -> (matrix, tile-row r, tile-col c) ----
    const int bz = blockIdx.x;
    const float* X; const float* Y;
    const float* sqx; const float* sqy;
    float* rowArr; float* colArr;
    bool doTrace = false;
    int r, c;
    if (bz < NTRI) {                       // AA, upper triangle
        int u = bz; r = 0;
        while (u >= NT - r) { u -= NT - r; ++r; }
        c = r + u;
        X = a + (size_t)r * TILE * DIMK;  Y = a + (size_t)c * TILE * DIMK;
        sqx = sq + r * TILE;              sqy = sq + c * TILE;
        rowArr = saa + r * TILE;
        colArr = (r != c) ? (saa + c * TILE) : nullptr;   // mirror by symmetry
    } else if (bz < 2 * NTRI) {            // BB, upper triangle
        int u = bz - NTRI; r = 0;
        while (u >= NT - r) { u -= NT - r; ++r; }
        c = r + u;
        X = b + (size_t)r * TILE * DIMK;  Y = b + (size_t)c * TILE * DIMK;
        sqx = sq + BATCH + r * TILE;      sqy = sq + BATCH + c * TILE;
        rowArr = sbb + r * TILE;
        colArr = (r != c) ? (sbb + c * TILE) : nullptr;
    } else {                               // AB, full grid
        int u = bz - 2 * NTRI;
        r = u >> 5;  c = u & 31;
        X = a + (size_t)r * TILE * DIMK;  Y = b + (size_t)c * TILE * DIMK;
        sqx = sq + r * TILE;              sqy = sq + BATCH + c * TILE;
        rowArr = sabr + r * TILE;
        colArr = sabc + c * TILE;
        doTrace = (r == c);
    }

    // ---- stage tiles into LDS via Tensor Data Mover (wave 0 issues DMA) ----
    __shared__ __align__(16) float Xs[TILE * LROW];
    __shared__ __align__(16) float Ys[TILE * LROW];
    if (threadIdx.x < 32) {
        tdm_load_tile(X, (unsigned)(uintptr_t)&Xs[0]);
        tdm_load_tile(Y, (unsigned)(uintptr_t)&Ys[0]);
        __builtin_amdgcn_s_wait_tensorcnt(0);   // wave 0 waits for both DMAs
    }
    __syncthreads();                            // release the other 7 waves

    // ---- WMMA compute: each wave owns a 16-row strip, sweeps 8 col subtiles
    const int wave = threadIdx.x >> 5;           // 0..7  -> row strip
    const int lane = threadIdx.x & 31;
    const int l15  = lane & 15;
    const int klo  = (lane >> 4) << 1;           // 0 or 2 (A/B f32 frag layout)
    const int moff = (lane >> 4) << 3;           // accumulator M offset: 0 or 8

    const float* xrow = Xs + (wave * 16 + l15) * LROW + klo;

    float xn[8];
    #pragma unroll
    for (int v = 0; v < 8; ++v) xn[v] = sqx[wave * 16 + moff + v];

    float rpart[8];
    #pragma unroll
    for (int v = 0; v < 8; ++v) rpart[v] = 0.0f;
    float tdiag = 0.0f;

    for (int js = 0; js < 8; ++js) {
        const float* yrow = Ys + (js * 16 + l15) * LROW + klo;
        v8f acc = {0.f, 0.f, 0.f, 0.f, 0.f, 0.f, 0.f, 0.f};
        #pragma unroll
        for (int k = 0; k < DIMK; k += 4) {
            v2f af = *(const v2f*)(xrow + k);
            v2f bf = *(const v2f*)(yrow + k);
            // D = A(16x4,f32) x B(4x16,f32) + C(16x16,f32)
            acc = __builtin_amdgcn_wmma_f32_16x16x4_f32(
                    false, af, false, bf, (short)0, acc, false, false);
        }
        const float yn = sqy[js * 16 + l15];
        float csum = 0.0f;
        #pragma unroll
        for (int v = 0; v < 8; ++v) {
            float d2  = fmaxf(xn[v] + yn - 2.0f * acc[v], 0.0f);
            // sim = 1/(1+sqrt(d2)) via v_sqrt_f32 + v_rcp_f32 (fast trans ops)
            float sim = __builtin_amdgcn_rcpf(1.0f + __builtin_amdgcn_sqrtf(d2));
            rpart[v] += sim;                       // row partial (fixed N/lane)
            csum     += sim;                       // column partial
            if (doTrace && wave == js) {
                // diagonal elements: (lane<8, v==lane) or (lane>=24, v==lane-24)
                if ((lane < 8 && v == lane) || (lane >= 24 && v == lane - 24))
                    tdiag += sim;
            }
        }
        if (colArr) {
            csum += __shfl_xor(csum, 16);          // combine M=0..7 and M=8..15
            if (lane < 16) atomicAdd(&colArr[js * 16 + lane], csum);
        }
    }

    // ---- flush row sums: butterfly over the 16 lanes of each half ----
    #pragma unroll
    for (int v = 0; v < 8; ++v) {
        float rv = rpart[v];
        rv += __shfl_xor(rv, 1);  rv += __shfl_xor(rv, 2);
        rv += __shfl_xor(rv, 4);  rv += __shfl_xor(rv, 8);
        if (lane == 0)  atomicAdd(&rowArr[wave * 16 + v], rv);
        if (lane == 16) atomicAdd(&rowArr[wave * 16 + 8 + v], rv);
    }
    if (doTrace) {
        tdiag += __shfl_xor(tdiag, 1);  tdiag += __shfl_xor(tdiag, 2);
        tdiag += __shfl_xor(tdiag, 4);  tdiag += __shfl_xor(tdiag, 8);
        tdiag += __shfl_xor(tdiag, 16);
        if (lane == 0) atomicAdd(trace, tdiag);
    }
}

// ---------------- final scalar reduction ----------------
__global__ void finalize_kernel(const float* __restrict__ saa,
                                const float* __restrict__ sbb,
                                const float* __restrict__ sabr,
                                const float* __restrict__ sabc,
                                const float* __restrict__ trace,
                                float* __restrict__ out) {
    __shared__ float red1[256];
    __shared__ float red2[256];
    float s1 = 0.0f, s2 = 0.0f;
    for (int i = threadIdx.x; i < BATCH; i += 256) {
        s1 += logf(sabr[i] + sbb[i]);   // logsumexp_1 numerator terms
        s2 += logf(saa[i] + sabc[i]);   // logsumexp_2 numerator terms
    }
    red1[threadIdx.x] = s1;
    red2[threadIdx.x] = s2;
    __syncthreads();
    for (int s = 128; s > 0; s >>= 1) {
        if (threadIdx.x < s) {
            red1[threadIdx.x] += red1[threadIdx.x + s];
            red2[threadIdx.x] += red2[threadIdx.x + s];
        }
        __syncthreads();
    }
    if (threadIdx.x == 0) {
        const float invB      = 1.0f / (float)BATCH;
        const float raw_unif  = (red1[0] + red2[0]) * invB;  // lse1 + lse2
        const float alignment = trace[0] * invB;
        out[0] = -(alignment - raw_unif * 0.5f);
    }
}

// ---------------------------------------------------------------------------
extern "C" void kernel_launch(void* const* d_in, const int* in_sizes, int n_in,
                              void* d_out, int out_size, void* d_ws, size_t ws_size,
                              hipStream_t stream) {
    (void)in_sizes; (void)n_in; (void)out_size; (void)ws_size;
    const float* feats = (const float*)d_in[0];
    float* out = (float*)d_out;

    float* ws    = (float*)d_ws;
    float* sq    = ws;                 // 8192  squared norms (a then b)
    float* saa   = ws + 8192;          // 4096  rowsum sim_aa
    float* sbb   = ws + 12288;         // 4096  rowsum sim_bb
    float* sabr  = ws + 16384;         // 4096  rowsum sim_ab
    float* sabc  = ws + 20480;         // 4096  colsum sim_ab
    float* trace = ws + 24576;         // 1

    zero_ws_kernel<<<64, 256, 0, stream>>>(saa, 4 * BATCH + 1);
    sqnorm_kernel<<<1024, 256, 0, stream>>>(feats, sq);
    tile_kernel<<<2 * NTRI + NT * NT, 256, 0, stream>>>(
        feats, sq, saa, sbb, sabr, sabc, trace);
    finalize_kernel<<<1, 256, 0, stream>>>(saa, sbb, sabr, sabc, trace, out);
}